// HyperbolicLoss_90177133346937
// MI455X (gfx1250) — compile-verified
//
#include <hip/hip_runtime.h>
#include <hip/hip_bf16.h>

// ---------------------------------------------------------------------------
// HyperbolicLoss on MI455X (gfx1250).
// Design: 51MB embedding table is L2-resident (192MB L2). Per wave, 16 pairs:
// gather 32 rows -> LDS (padded stride 132 dwords, conflict-free), then 32x
// chained V_WMMA_F32_16X16X4_F32 compute the full-f32 16x16 U*V^T; diagonal =
// the 16 pairwise dots. duv = uu + vv - 2*dot with precomputed node norms.
// Neg kernel processes NEG tiles per block so MRR ranks are computed in-LDS.
// ---------------------------------------------------------------------------

typedef float v2f __attribute__((ext_vector_type(2)));
typedef float v8f __attribute__((ext_vector_type(8)));

#define DIMS   128
#define ROWPAD 132          // dword stride per LDS row: bank = (4r+4t+c)%64 -> conflict-free
#define EPSV   1e-5f
#define MAXNEG 16

__device__ __forceinline__ float wave_reduce_sum(float x) {
#pragma unroll
  for (int off = 16; off >= 1; off >>= 1) x += __shfl_xor(x, off, 32);
  return x;
}

// Gather 16 u-rows (lanes 0..15) and 16 v-rows (lanes 16..31) into LDS.
__device__ __forceinline__ void load_tile_rows(const float* __restrict__ h,
                                               const int* __restrict__ srcIdx,
                                               const int* __restrict__ dstIdx,
                                               long base, long npairs,
                                               float* __restrict__ lds) {
  const int lane = threadIdx.x;
  const int p = lane & 15;
  const long pairIdx = base + p;
  int row = 0;
  if (pairIdx < npairs) row = (lane < 16) ? srcIdx[pairIdx] : dstIdx[pairIdx];
  const float4* g = (const float4*)(h + (long)row * DIMS);
  float4* l4 = (float4*)(lds + lane * ROWPAD);   // lane*528B, 16B aligned
#pragma unroll
  for (int j = 0; j < DIMS / 4; ++j) l4[j] = g[j];
}

// 32x V_WMMA_F32_16X16X4_F32 over K=128: acc = U(16x128) * V^T(128x16).
// A frag (16x4 f32): lane L<16 holds U[L][4t],U[L][4t+1]; lane L>=16 holds
// U[L-16][4t+2],U[L-16][4t+3].  B = V^T has the mirrored layout, so both
// fragments use identical per-lane addressing into the padded LDS rows.
__device__ __forceinline__ v8f tile_wmma(const float* __restrict__ lds) {
  const int lane = threadIdx.x;
  const int r = lane & 15;
  const int hk = lane >> 4;  // 0: k-pair {4t,4t+1}; 1: {4t+2,4t+3}
  const v2f* U2 = (const v2f*)(lds + r * ROWPAD);
  const v2f* V2 = (const v2f*)(lds + (16 + r) * ROWPAD);
  v8f acc = {};
#pragma unroll
  for (int t = 0; t < DIMS / 4; ++t) {
    v2f a = U2[2 * t + hk];
    v2f b = V2[2 * t + hk];
    acc = __builtin_amdgcn_wmma_f32_16x16x4_f32(false, a, false, b,
                                                (short)0, acc, false, false);
  }
  return acc;
}

// C/D layout: VGPR q, lanes 0..15 -> (M=q, N=lane); lanes 16..31 -> (M=q+8, N=lane-16).
// Diagonal (m,m): m<8 at lane m vgpr m; m>=8 at lane m+16 vgpr m-8.
// Deliver dot of pair p to lane p (p<16).
__device__ __forceinline__ float extract_diag(v8f acc) {
  const int lane = threadIdx.x;
  float dot = 0.f;
#pragma unroll
  for (int q = 0; q < 8; ++q) {
    float lo = __shfl(acc[q], q, 32);
    float hi = __shfl(acc[q], q + 24, 32);
    if (lane == q) dot = lo;
    if (lane == q + 8) dot = hi;
  }
  return dot;
}

__device__ __forceinline__ float poincare_score(float uu, float vv, float dot) {
  float duv   = fmaxf(uu + vv - 2.0f * dot, 0.0f);
  float alpha = fmaxf(1.0f - uu, EPSV);
  float beta  = fmaxf(1.0f - vv, EPSV);
  float gamma = fmaxf(1.0f + 2.0f * duv / (alpha * beta), 1.0f + EPSV);
  float d = acoshf(gamma);
  return d * d;
}

// ---------------------------------------------------------------------------

__global__ void hl_init_kernel(double* acc) {
  if (threadIdx.x == 0) { acc[0] = 0.0; acc[1] = 0.0; acc[2] = 0.0; }
}

__global__ void hl_sqnorm_kernel(const float* __restrict__ h,
                                 float* __restrict__ sqn, int N) {
  int i = blockIdx.x * blockDim.x + threadIdx.x;
  if (i >= N) return;
  const float4* r = (const float4*)(h + (long)i * DIMS);
  float s = 0.f;
#pragma unroll
  for (int j = 0; j < DIMS / 4; ++j) {
    float4 q = r[j];
    s += q.x * q.x + q.y * q.y + q.z * q.z + q.w * q.w;
  }
  sqn[i] = s;
}

__global__ __launch_bounds__(32) void hl_pos_kernel(
    const float* __restrict__ h, const int* __restrict__ src,
    const int* __restrict__ dst, const float* __restrict__ sqn,
    float* __restrict__ pos_score, double* acc, long npairs) {
  __shared__ float tile[32 * ROWPAD];
  const long base = (long)blockIdx.x * 16;
  load_tile_rows(h, src, dst, base, npairs, tile);
  __syncthreads();
  v8f accv = tile_wmma(tile);
  float dot = extract_diag(accv);

  const int lane = threadIdx.x;
  float contrib = 0.f;
  if (lane < 16) {
    long pi = base + lane;
    if (pi < npairs) {
      float uu = sqn[src[pi]];
      float vv = sqn[dst[pi]];
      float sc = poincare_score(uu, vv, dot);
      pos_score[pi] = sc;
      float sig = 1.0f / (1.0f + expf(-sc));
      contrib = logf(1.0f - sig + 1e-5f);
    }
  }
  float s = wave_reduce_sum(contrib);
  if (lane == 0) atomicAdd(&acc[0], (double)s);
}

// One block handles 16 pos groups = 16*negk negs (negk tiles of 16 pairs).
__global__ __launch_bounds__(32) void hl_neg_kernel(
    const float* __restrict__ h, const int* __restrict__ src,
    const int* __restrict__ dst, const float* __restrict__ sqn,
    const float* __restrict__ pos_score, double* acc,
    long nneg, long npos, int negk) {
  __shared__ float tile[32 * ROWPAD];
  __shared__ float nsc[16 * MAXNEG];
  const int lane = threadIdx.x;
  const long blockBase = (long)blockIdx.x * 16 * negk;
  float lsum = 0.f;

  for (int t = 0; t < negk; ++t) {
    long base = blockBase + (long)t * 16;
    load_tile_rows(h, src, dst, base, nneg, tile);
    __syncthreads();
    v8f accv = tile_wmma(tile);
    float dot = extract_diag(accv);
    if (lane < 16) {
      long pi = base + lane;
      if (pi < nneg) {
        float uu = sqn[src[pi]];
        float vv = sqn[dst[pi]];
        float sc = poincare_score(uu, vv, dot);
        nsc[t * 16 + lane] = sc;
        float sig = 1.0f / (1.0f + expf(-sc));
        lsum += logf(sig + 1e-5f);
      }
    }
    __syncthreads();
  }
  float s = wave_reduce_sum(lsum);
  if (lane == 0) atomicAdd(&acc[1], (double)s);

  // MRR: lane p owns pos group (blockIdx*16 + p); its negs sit at nsc[p*negk..].
  float mc = 0.f;
  long posIdx = (long)blockIdx.x * 16 + lane;
  if (lane < 16 && posIdx < npos) {
    float ps = pos_score[posIdx];
    int cnt = 0;
    for (int k = 0; k < negk; ++k) cnt += (nsc[lane * negk + k] <= ps) ? 1 : 0;
    mc = 1.0f / (float)(cnt + 1);
  }
  float ms = wave_reduce_sum(mc);
  if (lane == 0) atomicAdd(&acc[2], (double)ms);
}

__global__ void hl_final_kernel(const double* acc, float* out, long npos, long nneg) {
  if (threadIdx.x == 0) {
    out[0] = (float)(-(acc[0] / (double)npos) - (acc[1] / (double)nneg));
    out[1] = (float)(acc[2] / (double)npos);
  }
}

// ---------------------------------------------------------------------------

extern "C" void kernel_launch(void* const* d_in, const int* in_sizes, int n_in,
                              void* d_out, int out_size, void* d_ws, size_t ws_size,
                              hipStream_t stream) {
  const float* h       = (const float*)d_in[0];
  const int*   pos_src = (const int*)d_in[1];
  const int*   pos_dst = (const int*)d_in[2];
  const int*   neg_src = (const int*)d_in[3];
  const int*   neg_dst = (const int*)d_in[4];

  const long npos = in_sizes[1];
  const long nneg = in_sizes[3];
  int negk = (int)(nneg / (npos > 0 ? npos : 1));
  if (negk < 1) negk = 1;
  if (negk > MAXNEG) negk = MAXNEG;
  const int N = in_sizes[0] / DIMS;

  char* ws = (char*)d_ws;
  double* acc = (double*)ws;                                   // 3 f64 accumulators
  float* sqn = (float*)(ws + 64);                              // N floats
  size_t off = 64 + (((size_t)N * 4 + 63) / 64) * 64;
  float* pos_score = (float*)(ws + off);                       // npos floats

  hl_init_kernel<<<1, 32, 0, stream>>>(acc);
  hl_sqnorm_kernel<<<(N + 255) / 256, 256, 0, stream>>>(h, sqn, N);

  const unsigned posBlocks = (unsigned)((npos + 15) / 16);
  hl_pos_kernel<<<posBlocks, 32, 0, stream>>>(h, pos_src, pos_dst, sqn,
                                              pos_score, acc, npos);

  const long negPerBlock = 16L * negk;
  const unsigned negBlocks = (unsigned)((nneg + negPerBlock - 1) / negPerBlock);
  hl_neg_kernel<<<negBlocks, 32, 0, stream>>>(h, neg_src, neg_dst, sqn,
                                              pos_score, acc, nneg, npos, negk);

  hl_final_kernel<<<1, 32, 0, stream>>>(acc, (float*)d_out, npos, nneg);
}